// GaussianRenderer_41918880809704
// MI455X (gfx1250) — compile-verified
//
#include <hip/hip_runtime.h>

#define N_G   512
#define IMG_H 256
#define IMG_W 256
#define EPS_C 1e-4f
#define TWO_PI 6.283185307179586f

#define GLOBAL_AS __attribute__((address_space(1)))
#define LDS_AS    __attribute__((address_space(3)))

typedef int v4i __attribute__((ext_vector_type(4)));

// ---------------------------------------------------------------------------
// Kernel 1: per-gaussian preprocess + stable depth sort (one block, 512 thr).
// Emits a packed table (3 x float4 per gaussian, depth-sorted):
//   rec[0] = {mean2d.x, mean2d.y, ia, ib+ic}
//   rec[1] = {idd, coef (=valid*opac/(2*pi*det)), color.r, color.g}
//   rec[2] = {color.b, depth, 0, 0}
// ---------------------------------------------------------------------------
__global__ __launch_bounds__(N_G) void preprocess_sort_kernel(
    const float* __restrict__ means3D,   // N,3
    const float* __restrict__ covs3d,    // N,3,3
    const float* __restrict__ colors,    // N,3
    const float* __restrict__ opac,      // N
    const float* __restrict__ Km,        // 3,3
    const float* __restrict__ Rm,        // 3,3
    const float* __restrict__ tv,        // 3
    float4* __restrict__ table)          // N*3 float4 (sorted output)
{
    __shared__ float s_depth[N_G];
    const int i = threadIdx.x;

    // Uniform camera params (scalar loads)
    float Rl[3][3];
#pragma unroll
    for (int r = 0; r < 3; ++r)
#pragma unroll
        for (int c = 0; c < 3; ++c)
            Rl[r][c] = Rm[r * 3 + c];
    const float t0 = tv[0], t1 = tv[1], t2 = tv[2];
    const float fx = Km[0], cx = Km[2], fy = Km[4], cy = Km[5];

    const float mx3 = means3D[i * 3 + 0];
    const float my3 = means3D[i * 3 + 1];
    const float mz3 = means3D[i * 3 + 2];

    // cam = R*m + t
    const float X = Rl[0][0] * mx3 + Rl[0][1] * my3 + Rl[0][2] * mz3 + t0;
    const float Y = Rl[1][0] * mx3 + Rl[1][1] * my3 + Rl[1][2] * mz3 + t1;
    const float Z = Rl[2][0] * mx3 + Rl[2][1] * my3 + Rl[2][2] * mz3 + t2;

    const float depth = fmaxf(Z, 1.0f);

    // screen = K*cam; means2D = screen.xy / screen.z  (un-clipped Z, as ref)
    const float invsz = 1.0f / Z;
    const float m2x = (fx * X + cx * Z) * invsz;
    const float m2y = (fy * Y + cy * Z) * invsz;

    // J (2x3) with z = cam.z (un-clipped), world-space mean (as ref)
    const float invz = 1.0f / Z;
    const float J00 = fx * invz;
    const float J02 = -fx * mx3 * invz * invz;
    const float J11 = fy * invz;
    const float J12 = -fy * my3 * invz * invz;

    // Ccam = R * C * R^T
    float C[3][3];
#pragma unroll
    for (int r = 0; r < 3; ++r)
#pragma unroll
        for (int c = 0; c < 3; ++c)
            C[r][c] = covs3d[i * 9 + r * 3 + c];

    float RC[3][3], CC[3][3];
#pragma unroll
    for (int r = 0; r < 3; ++r)
#pragma unroll
        for (int c = 0; c < 3; ++c)
            RC[r][c] = Rl[r][0] * C[0][c] + Rl[r][1] * C[1][c] + Rl[r][2] * C[2][c];
#pragma unroll
    for (int r = 0; r < 3; ++r)
#pragma unroll
        for (int c = 0; c < 3; ++c)
            CC[r][c] = RC[r][0] * Rl[c][0] + RC[r][1] * Rl[c][1] + RC[r][2] * Rl[c][2];

    // cov2D = J * CC * J^T + EPS*I   (J row0=(J00,0,J02), row1=(0,J11,J12))
    const float M00 = J00 * CC[0][0] + J02 * CC[2][0];
    const float M01 = J00 * CC[0][1] + J02 * CC[2][1];
    const float M02 = J00 * CC[0][2] + J02 * CC[2][2];
    const float M10 = J11 * CC[1][0] + J12 * CC[2][0];
    const float M11 = J11 * CC[1][1] + J12 * CC[2][1];
    const float M12 = J11 * CC[1][2] + J12 * CC[2][2];

    const float a = M00 * J00 + M02 * J02 + EPS_C;
    const float b = M01 * J11 + M02 * J12;
    const float c = M10 * J00 + M12 * J02;
    const float d = M11 * J11 + M12 * J12 + EPS_C;

    const float det    = a * d - b * c;
    const float invdet = 1.0f / det;
    const float ia  = d * invdet;
    const float ibc = -(b + c) * invdet;   // ib + ic
    const float idd = a * invdet;

    const bool  valid = (depth > 1.0f) && (depth < 50.0f);
    const float coef  = valid ? (opac[i] / (TWO_PI * det)) : 0.0f;

    // ---- stable sort by depth: O(N) rank per thread via LDS ----
    s_depth[i] = depth;
    __syncthreads();
    int rank = 0;
    const float di = depth;
#pragma unroll 8
    for (int j = 0; j < N_G; ++j) {
        const float dj = s_depth[j];
        rank += (dj < di) || (dj == di && j < i);
    }

    const float cr = colors[i * 3 + 0];
    const float cg = colors[i * 3 + 1];
    const float cb = colors[i * 3 + 2];
    table[rank * 3 + 0] = make_float4(m2x, m2y, ia, ibc);
    table[rank * 3 + 1] = make_float4(idd, coef, cr, cg);
    table[rank * 3 + 2] = make_float4(cb, di, 0.0f, 0.0f);
}

// ---------------------------------------------------------------------------
// Kernel 2: per-pixel front-to-back compositing. 16x16 pixel tile / block.
// The 24KB sorted-gaussian table is staged into LDS with CDNA5 async
// global->LDS loads (ASYNCcnt), then read via broadcast ds_load_b128.
// ---------------------------------------------------------------------------
__global__ __launch_bounds__(256) void render_kernel(
    const float4* __restrict__ table,    // N_G*3 float4, depth-sorted
    float* __restrict__ out)             // H,W,3
{
    __shared__ float4 s_tbl[N_G * 3];    // 24 KB

    const int t = threadIdx.x;

    // Async-stage the table: 1536 float4 / 256 threads = 6 per thread.
#pragma unroll
    for (int k = 0; k < 6; ++k) {
        const int idx = t + k * 256;
        __builtin_amdgcn_global_load_async_to_lds_b128(
            (GLOBAL_AS v4i*)(table + idx),
            (LDS_AS v4i*)(s_tbl + idx),
            0, 0);
    }
    asm volatile("s_wait_asynccnt 0" ::: "memory");
    __syncthreads();

    const int x = blockIdx.x * 16 + (t & 15);
    const int y = blockIdx.y * 16 + (t >> 4);
    const float px = (float)x;
    const float py = (float)y;
    const bool pix0 = (x == 0) && (y == 0);   // ref's weights.at[:,0] quirk

    // Gaussian 0: weight uses the FULL transmittance (ref's roll-wrap quirk)
    float4 v0 = s_tbl[0], v1 = s_tbl[1], v2 = s_tbl[2];
    float dx = px - v0.x, dy = py - v0.y;
    float e  = v0.z * dx * dx + v0.w * dx * dy + v1.x * dy * dy;
    const float af0 = v1.y * __expf(-0.5f * e);
    const float c0r = v1.z, c0g = v1.w, c0b = v2.x;

    float T = 1.0f - af0 + 1e-10f;
    float r = 0.0f, g = 0.0f, b = 0.0f;

    for (int i = 1; i < N_G; ++i) {
        v0 = s_tbl[i * 3 + 0];
        v1 = s_tbl[i * 3 + 1];
        v2 = s_tbl[i * 3 + 2];
        dx = px - v0.x;
        dy = py - v0.y;
        e  = v0.z * dx * dx + v0.w * dx * dy + v1.x * dy * dy;
        const float alpha = v1.y * __expf(-0.5f * e);
        const float w = alpha * (pix0 ? 1.0f : T);
        r = fmaf(w, v1.z, r);
        g = fmaf(w, v1.w, g);
        b = fmaf(w, v2.x, b);
        T = T * (1.0f - alpha + 1e-10f);
    }

    const float w0 = af0 * (pix0 ? 1.0f : T);  // T is now the full product
    r = fmaf(w0, c0r, r);
    g = fmaf(w0, c0g, g);
    b = fmaf(w0, c0b, b);

    const int o = (y * IMG_W + x) * 3;
    out[o + 0] = r;
    out[o + 1] = g;
    out[o + 2] = b;
}

// ---------------------------------------------------------------------------
extern "C" void kernel_launch(void* const* d_in, const int* in_sizes, int n_in,
                              void* d_out, int out_size, void* d_ws, size_t ws_size,
                              hipStream_t stream) {
    const float* means3D   = (const float*)d_in[0];  // N,3
    const float* covs3d    = (const float*)d_in[1];  // N,3,3
    const float* colors    = (const float*)d_in[2];  // N,3
    const float* opacities = (const float*)d_in[3];  // N
    const float* Km        = (const float*)d_in[4];  // 3,3
    const float* Rm        = (const float*)d_in[5];  // 3,3
    const float* tv        = (const float*)d_in[6];  // 3
    // d_in[7] = pixels (H,W,2) — coordinates synthesized on-chip instead.

    float4* table = (float4*)d_ws;                   // N_G*3 float4 = 24 KB
    float*  out   = (float*)d_out;                   // H,W,3

    preprocess_sort_kernel<<<1, N_G, 0, stream>>>(
        means3D, covs3d, colors, opacities, Km, Rm, tv, table);

    dim3 grid(IMG_W / 16, IMG_H / 16);
    render_kernel<<<grid, 256, 0, stream>>>(table, out);
}